// GCN_18442589569633
// MI455X (gfx1250) — compile-verified
//
#include <hip/hip_runtime.h>
#include <hip/hip_bf16.h>

#define N_NODES   50000
#define N_EDGES   600000
#define BAG       16
#define EMB       128
#define HID       128
#define N_CLASSES 10
#define N_GRAPHS  512
#define PAD_IDX   1

typedef __attribute__((ext_vector_type(2))) float v2f;
typedef __attribute__((ext_vector_type(8))) float v8f;

// ---------------------------------------------------------------------------
// Zero-fill (grid-stride)
// ---------------------------------------------------------------------------
__global__ void zero_f32(float* __restrict__ p, int n) {
    int i = blockIdx.x * blockDim.x + threadIdx.x;
    for (; i < n; i += gridDim.x * blockDim.x) p[i] = 0.0f;
}

// ---------------------------------------------------------------------------
// EmbeddingBag(mode='mean', padding_idx=1): one wave per node, lane owns 4
// consecutive features (128 / 32 = 4) -> one float4 gather per token.
// ---------------------------------------------------------------------------
__global__ void emb_bag_mean(const int* __restrict__ tokens,
                             const float* __restrict__ table,
                             float* __restrict__ out) {
    const int node = blockIdx.x * (blockDim.x >> 5) + (threadIdx.x >> 5);
    if (node >= N_NODES) return;
    const int lane = threadIdx.x & 31;
    const int* t = tokens + node * BAG;
    float4 acc = make_float4(0.f, 0.f, 0.f, 0.f);
    float cnt = 0.f;
    #pragma unroll
    for (int i = 0; i < BAG; ++i) {
        const int tok = t[i];
        if (tok != PAD_IDX) {
            const float4 e = *(const float4*)(table + (size_t)tok * EMB + lane * 4);
            acc.x += e.x; acc.y += e.y; acc.z += e.z; acc.w += e.w;
            cnt += 1.0f;
        }
    }
    const float inv = 1.0f / fmaxf(cnt, 1.0f);
    float4 r = make_float4(acc.x * inv, acc.y * inv, acc.z * inv, acc.w * inv);
    *(float4*)(out + (size_t)node * EMB + lane * 4) = r;
}

// ---------------------------------------------------------------------------
// Degree / graph-size histograms
// ---------------------------------------------------------------------------
__global__ void count_deg(const int* __restrict__ dst, float* __restrict__ deg) {
    const int e = blockIdx.x * blockDim.x + threadIdx.x;
    if (e < N_EDGES) atomicAdd(&deg[dst[e]], 1.0f);
}

__global__ void count_graph(const int* __restrict__ batch, float* __restrict__ gcnt) {
    const int i = blockIdx.x * blockDim.x + threadIdx.x;
    if (i < N_NODES) atomicAdd(&gcnt[batch[i]], 1.0f);
}

// ---------------------------------------------------------------------------
// Edge scatter: msg[dst] += x[src]. One wave per edge; lane owns 4 floats.
// x (25.6 MB) is L2-resident on MI455X (192 MB L2), so this is L2-bandwidth
// bound, not HBM bound.
// ---------------------------------------------------------------------------
__global__ void edge_scatter(const int* __restrict__ src,
                             const int* __restrict__ dst,
                             const float* __restrict__ x,
                             float* __restrict__ msg) {
    const int e = blockIdx.x * (blockDim.x >> 5) + (threadIdx.x >> 5);
    if (e >= N_EDGES) return;
    const int lane = threadIdx.x & 31;
    const int s = src[e];
    const int d = dst[e];
    const float4 v = *(const float4*)(x + (size_t)s * HID + lane * 4);
    float* p = msg + (size_t)d * HID + lane * 4;
    atomicAdd(p + 0, v.x);
    atomicAdd(p + 1, v.y);
    atomicAdd(p + 2, v.z);
    atomicAdd(p + 3, v.w);
}

// ---------------------------------------------------------------------------
// Fused SAGEConv update:
//   out = relu( (msg/deg) @ wl^T + b + x @ wr^T )
// One wave computes one 16x16 output tile with V_WMMA_F32_16X16X4_F32.
// A-frag (16x4 f32): lanes 0-15 -> M=lane, K={0,1}; lanes 16-31 -> same M,
// K={2,3}; so each lane holds a fixed row -> fold 1/deg as a lane scalar.
// B-frag (4x16 f32) mirrors A with N=lane%16.  B[k][n] = W[n][k] (W row-major
// [HID, HID]) so a float2 load of W[n] at offset k+2*half supplies the frag.
// C/D: VGPR r -> M = r + 8*(lane/16), N = lane%16; bias b[n] is a lane scalar.
// N_NODES % 16 == 0, so EXEC is all-1s throughout (WMMA requirement).
// ---------------------------------------------------------------------------
__global__ void sage_wmma(const float* __restrict__ msg,
                          const float* __restrict__ deg,
                          const float* __restrict__ x,
                          const float* __restrict__ wl,
                          const float* __restrict__ bias,
                          const float* __restrict__ wr,
                          float* __restrict__ out) {
    const int lane   = threadIdx.x & 31;
    const int wave   = threadIdx.x >> 5;
    const int half   = lane >> 4;       // 0: K={0,1} / M rows 0-7 ; 1: K={2,3} / M rows 8-15
    const int lrow   = lane & 15;
    const int m_base = blockIdx.x * 16;
    const int n_base = (blockIdx.y * 4 + wave) * 16;

    const int arow = m_base + lrow;                 // per-lane fixed A row
    const int brow = n_base + lrow;                 // per-lane fixed W row (output feature)
    const float rdeg = 1.0f / fmaxf(deg[arow], 1.0f);

    v8f c;
    const float bv = bias[brow];
    #pragma unroll
    for (int r = 0; r < 8; ++r) c[r] = bv;

    const float* mrow = msg + (size_t)arow * HID;
    const float* xrow = x   + (size_t)arow * HID;
    const float* lrowp = wl + (size_t)brow * HID;
    const float* rrowp = wr + (size_t)brow * HID;

    #pragma unroll 8
    for (int k = 0; k < HID; k += 4) {
        const int ka = k + half * 2;                // even -> 8B aligned float2
        v2f am = *(const v2f*)(mrow + ka);
        am *= rdeg;                                 // aggr = msg / clip(deg,1)
        const v2f ax = *(const v2f*)(xrow + ka);
        const v2f bl = *(const v2f*)(lrowp + ka);
        const v2f br = *(const v2f*)(rrowp + ka);
        c = __builtin_amdgcn_wmma_f32_16x16x4_f32(false, am, false, bl,
                                                  (short)0, c, false, false);
        c = __builtin_amdgcn_wmma_f32_16x16x4_f32(false, ax, false, br,
                                                  (short)0, c, false, false);
    }

    #pragma unroll
    for (int r = 0; r < 8; ++r) {
        const float v = fmaxf(c[r], 0.0f);          // relu
        out[(size_t)(m_base + half * 8 + r) * HID + n_base + lrow] = v;
    }
}

// ---------------------------------------------------------------------------
// global_mean_pool scatter: gsum[batch[node]] += x[node]
// ---------------------------------------------------------------------------
__global__ void pool_scatter(const float* __restrict__ x,
                             const int* __restrict__ batch,
                             float* __restrict__ gsum) {
    const int node = blockIdx.x * (blockDim.x >> 5) + (threadIdx.x >> 5);
    if (node >= N_NODES) return;
    const int lane = threadIdx.x & 31;
    const int g = batch[node];
    const float4 v = *(const float4*)(x + (size_t)node * HID + lane * 4);
    float* p = gsum + (size_t)g * HID + lane * 4;
    atomicAdd(p + 0, v.x);
    atomicAdd(p + 1, v.y);
    atomicAdd(p + 2, v.z);
    atomicAdd(p + 3, v.w);
}

// ---------------------------------------------------------------------------
// Output head: out[g][c] = (gsum[g]/cnt[g]) . w_out[c] + b_out[c]
// 5120 dot products of length 128 -> VALU is fine (negligible work).
// ---------------------------------------------------------------------------
__global__ void out_head(const float* __restrict__ gsum,
                         const float* __restrict__ gcnt,
                         const float* __restrict__ wout,
                         const float* __restrict__ bout,
                         float* __restrict__ out) {
    const int idx = blockIdx.x * blockDim.x + threadIdx.x;
    if (idx >= N_GRAPHS * N_CLASSES) return;
    const int g = idx / N_CLASSES;
    const int c = idx % N_CLASSES;
    const float inv = 1.0f / fmaxf(gcnt[g], 1.0f);
    const float* gv = gsum + (size_t)g * HID;
    const float* wv = wout + (size_t)c * HID;
    float s = 0.0f;
    #pragma unroll 8
    for (int k = 0; k < HID; ++k) s += gv[k] * wv[k];
    out[idx] = s * inv + bout[c];
}

// ---------------------------------------------------------------------------
// Launch
// ---------------------------------------------------------------------------
extern "C" void kernel_launch(void* const* d_in, const int* in_sizes, int n_in,
                              void* d_out, int out_size, void* d_ws, size_t ws_size,
                              hipStream_t stream) {
    (void)in_sizes; (void)n_in; (void)out_size; (void)ws_size;

    const int*   x_tokens = (const int*)  d_in[0];
    const int*   edge     = (const int*)  d_in[1];
    const int*   batch    = (const int*)  d_in[2];
    const float* emb      = (const float*)d_in[3];
    const float* w1l      = (const float*)d_in[4];
    const float* b1       = (const float*)d_in[5];
    const float* w1r      = (const float*)d_in[6];
    const float* w2l      = (const float*)d_in[7];
    const float* b2       = (const float*)d_in[8];
    const float* w2r      = (const float*)d_in[9];
    const float* wout     = (const float*)d_in[10];
    const float* bout     = (const float*)d_in[11];
    float*       out      = (float*)d_out;

    const int* src = edge;              // edge_index[0]
    const int* dst = edge + N_EDGES;    // edge_index[1]

    const size_t nodeF = (size_t)N_NODES * HID;   // 6,400,000 floats
    float* bufA = (float*)d_ws;                   // node features (current x)
    float* bufB = bufA + nodeF;                   // msg accumulator
    float* bufC = bufB + nodeF;                   // next-layer features
    float* deg  = bufC + nodeF;                   // [N_NODES]
    float* gsum = deg + N_NODES;                  // [N_GRAPHS * HID]
    float* gcnt = gsum + (size_t)N_GRAPHS * HID;  // [N_GRAPHS]

    const dim3 blk256(256);
    const dim3 gWave((N_NODES + 7) / 8);          // 8 waves/block, 1 wave/node
    const dim3 gEdge((N_EDGES + 7) / 8);          // 1 wave/edge
    const dim3 gGemm(N_NODES / 16, 2);            // 16-row tiles x (2*4 waves = 8 N-tiles)
    const dim3 bGemm(128);                        // 4 waves/block

    // Zero accumulators (msg + deg + gsum + gcnt are contiguous except bufC)
    zero_f32<<<2048, blk256, 0, stream>>>(bufB, (int)nodeF);
    zero_f32<<<256,  blk256, 0, stream>>>(deg, N_NODES);
    zero_f32<<<256,  blk256, 0, stream>>>(gsum, N_GRAPHS * HID);
    zero_f32<<<4,    blk256, 0, stream>>>(gcnt, N_GRAPHS);

    // x0 = embedding_bag_mean(tokens)
    emb_bag_mean<<<gWave, blk256, 0, stream>>>(x_tokens, emb, bufA);

    // in-degrees and graph sizes (reused by both convs / pooling)
    count_deg<<<(N_EDGES + 255) / 256, blk256, 0, stream>>>(dst, deg);
    count_graph<<<(N_NODES + 255) / 256, blk256, 0, stream>>>(batch, gcnt);

    // ---- SAGE layer 1 ----
    edge_scatter<<<gEdge, blk256, 0, stream>>>(src, dst, bufA, bufB);
    sage_wmma<<<gGemm, bGemm, 0, stream>>>(bufB, deg, bufA, w1l, b1, w1r, bufC);

    // ---- SAGE layer 2 ----
    zero_f32<<<2048, blk256, 0, stream>>>(bufB, (int)nodeF);
    edge_scatter<<<gEdge, blk256, 0, stream>>>(src, dst, bufC, bufB);
    sage_wmma<<<gGemm, bGemm, 0, stream>>>(bufB, deg, bufC, w2l, b2, w2r, bufA);

    // ---- global mean pool + classifier head ----
    pool_scatter<<<gWave, blk256, 0, stream>>>(bufA, batch, gsum);
    out_head<<<(N_GRAPHS * N_CLASSES + 255) / 256, blk256, 0, stream>>>(
        gsum, gcnt, wout, bout, out);
}